// ModelWithLoss_18167711662317
// MI455X (gfx1250) — compile-verified
//
#include <hip/hip_runtime.h>
#include <hip/hip_bf16.h>

#define A_CNT 49104
#define C_CNT 80
#define M_CNT 32
#define B_CNT 8
#define FL_EPS 1.0e-4f

typedef __attribute__((ext_vector_type(2))) float v2f;
typedef __attribute__((ext_vector_type(8))) float v8f;

// Exact f32 wave(32)-level sum using V_WMMA_F32_16X16X4_F32.
// A (16x4 f32): lane L<16 holds (A[L][0], A[L][1]); lane L>=16 holds (A[L-16][2], A[L-16][3]).
// We place the lane's partial in A[.][0]/A[.][2] and zero in the K=1/3 slots.
// B = all ones (4x16)  =>  D[m][n] = v(lane m) + v(lane m+16), broadcast over n.
// C/D layout: lane l, vgpr i -> element (M = i + 8*(l>=16), N = l%16), so summing the
// 8 accumulator components gives half-sums; one shfl_xor(16) completes the reduction.
__device__ __forceinline__ float wave_reduce_sum(float v) {
  v2f a; a.x = v;    a.y = 0.0f;
  v2f b; b.x = 1.0f; b.y = 1.0f;
  v8f c = {};
  c = __builtin_amdgcn_wmma_f32_16x16x4_f32(
      /*neg_a=*/false, a, /*neg_b=*/false, b,
      /*c_mod=*/(short)0, c, /*reuse_a=*/false, /*reuse_b=*/false);
  float s = c[0] + c[1] + c[2] + c[3] + c[4] + c[5] + c[6] + c[7];
  s += __shfl_xor(s, 16, 32);
  return s;  // every lane holds the full 32-lane sum
}

__global__ void fl_zero_ws(float* acc) {
  if (threadIdx.x < 3 * B_CNT) acc[threadIdx.x] = 0.0f;
}

__global__ void __launch_bounds__(256)
fl_main(const float* __restrict__ cls, const float* __restrict__ reg,
        const float* __restrict__ anc, const float* __restrict__ ann,
        float* __restrict__ acc) {
  const int b = blockIdx.y;
  const int a = blockIdx.x * blockDim.x + threadIdx.x;
  const bool valid = (a < A_CNT);
  const int aa = valid ? a : 0;  // safe clamped index; contributions gated by `valid`

  __shared__ float s_ann[M_CNT * 5];
  __shared__ float s_area[M_CNT];
  if (threadIdx.x < M_CNT * 5)
    s_ann[threadIdx.x] = ann[(size_t)b * (M_CNT * 5) + threadIdx.x];
  __syncthreads();
  if (threadIdx.x < M_CNT) {
    const float* bx = &s_ann[threadIdx.x * 5];
    s_area[threadIdx.x] = (bx[2] - bx[0]) * (bx[3] - bx[1]);  // (x2-x1)*(y2-y1)
  }
  __syncthreads();

  // Anchor row: [y1, x1, y2, x2], contiguous & 16B aligned.
  const float4 an = ((const float4*)anc)[aa];
  const float ya1 = an.x, xa1 = an.y, ya2 = an.z, xa2 = an.w;
  const float a_w = xa2 - xa1, a_h = ya2 - ya1;
  const float a_cx = xa1 + 0.5f * a_w, a_cy = ya1 + 0.5f * a_h;
  const float area_a = a_w * a_h;

  const float* cls_row = cls + ((size_t)b * A_CNT + (size_t)aa) * C_CNT;
  __builtin_prefetch(cls_row, 0, 1);  // global_prefetch_b8: overlap HBM fetch with IoU math

  // IoU max / argmax over the 32 GT boxes (first-max semantics like jnp.argmax).
  float iou_max = -1.0f;
  int iou_arg = 0;
  for (int m = 0; m < M_CNT; ++m) {
    const float* bx = &s_ann[m * 5];  // [x1, y1, x2, y2, label]
    float iw = fminf(xa2, bx[2]) - fmaxf(xa1, bx[0]);
    float ih = fminf(ya2, bx[3]) - fmaxf(ya1, bx[1]);
    iw = fmaxf(iw, 0.0f);
    ih = fmaxf(ih, 0.0f);
    const float inter = iw * ih;
    const float iou = inter / (area_a + s_area[m] - inter);
    if (iou > iou_max) { iou_max = iou; iou_arg = m; }
  }

  const bool pos = valid && (iou_max >= 0.5f);
  const bool neg = valid && (iou_max < 0.4f);

  // Classification focal loss (skipped entirely for the ignore band 0.4 <= iou < 0.5).
  float cls_sum = 0.0f;
  if (pos || neg) {
    const int label = pos ? (int)s_ann[iou_arg * 5 + 4] : -1;
    const float4* cp = (const float4*)cls_row;  // 80 f32 = 20 float4, 16B aligned
#pragma unroll 4
    for (int i = 0; i < C_CNT / 4; ++i) {
      const float4 p4 = cp[i];
      const float pv[4] = {p4.x, p4.y, p4.z, p4.w};
#pragma unroll
      for (int j = 0; j < 4; ++j) {
        const float p = fminf(fmaxf(pv[j], FL_EPS), 1.0f - FL_EPS);
        const bool t1 = ((4 * i + j) == label);
        const float q = t1 ? (1.0f - p) : p;           // focal weight base
        const float fw = (t1 ? 0.25f : 0.75f) * q * q; // alpha * q^2
        const float bce = -__logf(t1 ? p : (1.0f - p));
        cls_sum += fw * bce;
      }
    }
  }

  // Smooth-L1 regression loss for positive anchors.
  float reg_sum = 0.0f;
  if (pos) {
    const float* bx = &s_ann[iou_arg * 5];
    const float gx1 = bx[0], gy1 = bx[1], gx2 = bx[2], gy2 = bx[3];
    float g_w = gx2 - gx1, g_h = gy2 - gy1;
    const float g_cx = gx1 + 0.5f * g_w;  // centers use *unclipped* w/h (matches ref)
    const float g_cy = gy1 + 0.5f * g_h;
    g_w = fmaxf(g_w, 1.0f);
    g_h = fmaxf(g_h, 1.0f);
    const float tdx = (g_cx - a_cx) / a_w;
    const float tdy = (g_cy - a_cy) / a_h;
    const float tdw = __logf(g_w / a_w);
    const float tdh = __logf(g_h / a_h);
    const float4 rg = ((const float4*)reg)[(size_t)b * A_CNT + (size_t)aa];
    const float t[4] = {tdy, tdx, tdh, tdw};
    const float r[4] = {rg.x, rg.y, rg.z, rg.w};
#pragma unroll
    for (int k = 0; k < 4; ++k) {
      const float d = fabsf(t[k] - r[k]);
      reg_sum += (d <= (1.0f / 9.0f)) ? 4.5f * d * d : (d - 1.0f / 18.0f);
    }
  }

  // Wave32 reductions: two exact-f32 WMMA sums + ballot/popc for the positive count.
  // Uniform control flow here (no early returns), so EXEC is all-1s for the WMMAs.
  const float w_cls = wave_reduce_sum(cls_sum);
  const float w_reg = wave_reduce_sum(reg_sum);
  const unsigned long long bal = __ballot(pos);
  const int w_pos = __popcll(bal);

  if ((threadIdx.x & 31) == 0) {
    atomicAdd(&acc[b * 3 + 0], w_cls);
    atomicAdd(&acc[b * 3 + 1], w_reg);
    atomicAdd(&acc[b * 3 + 2], (float)w_pos);  // exact: counts << 2^24
  }
}

__global__ void fl_finalize(const float* __restrict__ acc, float* __restrict__ out) {
  const int b = threadIdx.x;  // single wave of 32
  float cls_l = 0.0f, reg_l = 0.0f;
  if (b < B_CNT) {
    const float sc = acc[b * 3 + 0];
    const float sr = acc[b * 3 + 1];
    const float np = acc[b * 3 + 2];
    cls_l = sc / fmaxf(np, 1.0f);
    reg_l = (np > 0.0f) ? (sr / fmaxf(np * 4.0f, 1.0f)) : 0.0f;
  }
#pragma unroll
  for (int off = 4; off >= 1; off >>= 1) {
    cls_l += __shfl_down(cls_l, off, 32);
    reg_l += __shfl_down(reg_l, off, 32);
  }
  if (b == 0) {
    out[0] = cls_l * (1.0f / B_CNT);
    out[1] = reg_l * (1.0f / B_CNT);
  }
}

extern "C" void kernel_launch(void* const* d_in, const int* in_sizes, int n_in,
                              void* d_out, int out_size, void* d_ws, size_t ws_size,
                              hipStream_t stream) {
  const float* cls = (const float*)d_in[0];  // (8, 49104, 80)
  const float* reg = (const float*)d_in[1];  // (8, 49104, 4)
  const float* anc = (const float*)d_in[2];  // (1, 49104, 4)
  const float* ann = (const float*)d_in[3];  // (8, 32, 5)
  float* acc = (float*)d_ws;                 // 8 x {cls_sum, reg_sum, num_pos}
  float* out = (float*)d_out;                // {mean_cls_loss, mean_reg_loss}

  fl_zero_ws<<<1, 32, 0, stream>>>(acc);
  dim3 grid((A_CNT + 255) / 256, B_CNT);
  fl_main<<<grid, 256, 0, stream>>>(cls, reg, anc, ann, acc);
  fl_finalize<<<1, 32, 0, stream>>>(acc, out);
}